// MeshUnpool_14946486190524
// MI455X (gfx1250) — compile-verified
//
#include <hip/hip_runtime.h>
#include <stdint.h>
#include <stddef.h>

// Problem constants (match reference file)
#define BB      4
#define NFULL   65536
#define NSMALL  32768
#define KK      32768
#define FF      256

typedef __attribute__((ext_vector_type(4))) float f4;
typedef __attribute__((ext_vector_type(4))) int   i4;

#if __has_builtin(__builtin_amdgcn_global_load_async_to_lds_b128) && \
    __has_builtin(__builtin_amdgcn_global_store_async_from_lds_b128)
#define USE_ASYNC_LDS 1
#else
#define USE_ASYNC_LDS 0
#endif

__device__ __forceinline__ void wait_async0() {
#if __has_builtin(__builtin_amdgcn_s_wait_asynccnt)
  __builtin_amdgcn_s_wait_asynccnt(0);
#else
  asm volatile("s_wait_asynccnt 0" ::: "memory");
#endif
}

// ---------------- heavy gather, one wave32 per 1KB row (defined FIRST so the
// ---------------- disasm snippet shows its codegen) ----------------
__global__ void mu_gather_rows_kernel(const float* __restrict__ images,
                                      const int* __restrict__ srcPos,
                                      float* __restrict__ out) {
  __shared__ float lds[8 * FF];                         // 8 waves * 1KB
  const int wave = threadIdx.x >> 5;
  const int lane = threadIdx.x & 31;
  const int row  = blockIdx.x * 8 + wave;               // grid sized exactly: B*NFULL/8 blocks
  const int b    = row >> 16;
  const int p    = srcPos[row];
  float* orow = out + (size_t)row * FF;

  if (p < 0) {                                          // uniform per wave
    f4 z = {0.f, 0.f, 0.f, 0.f};
    ((f4*)orow)[lane]      = z;
    ((f4*)orow)[lane + 32] = z;
    return;
  }
  const float* irow = images + ((size_t)b * NSMALL + (size_t)p) * FF;

#if USE_ASYNC_LDS
  // Stage the 1KB row through LDS with the gfx1250 async DMA path
  // (ASYNCcnt-tracked, bypasses VGPRs). 32 lanes * 16B * 2 = 1KB.
  float* l0 = &lds[wave * FF + lane * 4];
  float* l1 = &lds[wave * FF + 128 + lane * 4];
  __builtin_amdgcn_global_load_async_to_lds_b128(
      (i4*)(irow + lane * 4),        (i4*)l0, 0, 0);
  __builtin_amdgcn_global_load_async_to_lds_b128(
      (i4*)(irow + 128 + lane * 4),  (i4*)l1, 0, 0);
  wait_async0();                                        // loads landed in LDS
  __builtin_amdgcn_global_store_async_from_lds_b128(
      (i4*)(orow + lane * 4),        (i4*)l0, 0, 0);
  __builtin_amdgcn_global_store_async_from_lds_b128(
      (i4*)(orow + 128 + lane * 4),  (i4*)l1, 0, 0);
  // no trailing wait: S_ENDPGM performs an implicit wait-idle; LDS slice is
  // private to this wave and not reused.
#else
  (void)lds;
  f4 a0 = ((const f4*)irow)[lane];
  f4 a1 = ((const f4*)irow)[lane + 32];
  ((f4*)orow)[lane]      = a0;
  ((f4*)orow)[lane + 32] = a1;
#endif
}

// ---------------- phase 1: init pos[] and head[] to -1 ----------------
__global__ void mu_init_kernel(int* __restrict__ w, int n) {
  int i = blockIdx.x * blockDim.x + threadIdx.x;
  if (i < n) w[i] = -1;
}

// ---------------- phase 2: pos[b][mask_idx[b][i]] = i ----------------
__global__ void mu_scatter_pos_kernel(const int* __restrict__ mask_idx,
                                      int* __restrict__ pos) {
  int idx = blockIdx.x * blockDim.x + threadIdx.x;      // [0, B*NSMALL)
  if (idx >= BB * NSMALL) return;
  int b = idx >> 15;                                    // / NSMALL
  int i = idx & (NSMALL - 1);
  int m = mask_idx[idx];                                // sorted unique per batch
  pos[b * NFULL + m] = i;
}

// ---------------- phase 3: per-target linked list of write times ----------------
// execution time j = 0..K-1 corresponds to reversed pair k = K-1-j
// op j: v[t_j] = v[f_j], t_j = order[b,1,K-1-j]
__global__ void mu_build_lists_kernel(const int* __restrict__ order,
                                      int* __restrict__ head,
                                      int* __restrict__ nextOp) {
  int idx = blockIdx.x * blockDim.x + threadIdx.x;      // [0, B*K)
  if (idx >= BB * KK) return;
  int b = idx >> 15;                                    // / KK
  int j = idx & (KK - 1);
  int k = KK - 1 - j;
  int t = order[(size_t)b * 2 * KK + KK + k];           // order[b,1,k]
  nextOp[idx] = atomicExch(&head[b * NFULL + t], j);
}

// ---------------- phase 4: resolve root(r) and its image row ----------------
__global__ void mu_resolve_kernel(const int* __restrict__ order,
                                  const int* __restrict__ head,
                                  const int* __restrict__ nextOp,
                                  const int* __restrict__ pos,
                                  int* __restrict__ srcPos) {
  int idx = blockIdx.x * blockDim.x + threadIdx.x;      // b*NFULL + r
  if (idx >= BB * NFULL) return;
  int b = idx >> 16;                                    // / NFULL
  int x = idx & (NFULL - 1);
  int T = KK;                                           // "before time T"
  for (;;) {
    int best = -1;
    for (int j = head[b * NFULL + x]; j >= 0; j = nextOp[b * KK + j]) {
      if (j < T && j > best) best = j;                  // latest write before T
    }
    if (best < 0) break;
    x = order[(size_t)b * 2 * KK + (KK - 1 - best)];    // f = order[b,0,K-1-best]
    T = best;                                           // strictly decreasing -> terminates
  }
  srcPos[idx] = pos[b * NFULL + x];                     // -1 => zero row
}

extern "C" void kernel_launch(void* const* d_in, const int* in_sizes, int n_in,
                              void* d_out, int out_size, void* d_ws, size_t ws_size,
                              hipStream_t stream) {
  (void)in_sizes; (void)n_in; (void)out_size; (void)ws_size;
  const float* images   = (const float*)d_in[0];  // [B, NSMALL, F]
  const int*   mask_idx = (const int*)d_in[1];    // [B, NSMALL]
  const int*   order    = (const int*)d_in[2];    // [B, 2, K]
  float*       out      = (float*)d_out;          // [B, NFULL, F]

  // workspace layout (ints): pos | head | nextOp | srcPos  (~3.5 MB)
  int* W      = (int*)d_ws;
  int* pos    = W;
  int* head   = W + BB * NFULL;
  int* nextOp = W + 2 * BB * NFULL;
  int* srcPos = W + 2 * BB * NFULL + BB * KK;

  const int initN = 2 * BB * NFULL;                // pos + head contiguous
  hipLaunchKernelGGL(mu_init_kernel, dim3((initN + 255) / 256), dim3(256), 0, stream,
                     W, initN);
  hipLaunchKernelGGL(mu_scatter_pos_kernel, dim3((BB * NSMALL + 255) / 256), dim3(256), 0, stream,
                     mask_idx, pos);
  hipLaunchKernelGGL(mu_build_lists_kernel, dim3((BB * KK + 255) / 256), dim3(256), 0, stream,
                     order, head, nextOp);
  hipLaunchKernelGGL(mu_resolve_kernel, dim3((BB * NFULL + 255) / 256), dim3(256), 0, stream,
                     order, head, nextOp, pos, srcPos);
  hipLaunchKernelGGL(mu_gather_rows_kernel, dim3(BB * NFULL / 8), dim3(256), 0, stream,
                     images, srcPos, out);
}